// SAGE_20572893348074
// MI455X (gfx1250) — compile-verified
//
#include <hip/hip_runtime.h>
#include <hip/hip_bf16.h>
#include <stdint.h>

// ---------------------------------------------------------------------------
// GraphSAGE 3-layer forward for MI455X (gfx1250).
//  - CSR built on device (histogram + 1-block scan + bucket) -> no fp atomics
//  - per-dst coalesced float4 gather-mean writes normalized h_neigh
//  - GEMMs via V_WMMA_F32_16X16X4_F32 (exact fp32 matrix pipe), fully
//    compile-time-specialized, pair-packed weights -> b64 B loads
// ---------------------------------------------------------------------------

typedef __attribute__((ext_vector_type(2))) float v2f;
typedef __attribute__((ext_vector_type(8))) float v8f;

#define N_SRC0 2162688
#define N_DST0 135168
#define N_E0   2027520
#define N_DST1 12288
#define N_E1   122880
#define N_DST2 2048
#define N_E2   10240
#define F_IN   128
#define F_HID  256
#define F_OUT  47
#define F_OUT2P 128   // layer-2 padded column count (8 waves x 16)

// ---------------------------- small utility kernels ------------------------

__global__ void zero_ints(int* __restrict__ p, int n) {
  int i = blockIdx.x * blockDim.x + threadIdx.x;
  if (i < n) p[i] = 0;
}

__global__ void count_edges(const int* __restrict__ dst, int* __restrict__ cnt, int nE) {
  int e = blockIdx.x * blockDim.x + threadIdx.x;
  if (e < nE) atomicAdd(&cnt[dst[e]], 1);
}

// single-block exclusive scan (n <= ~135k, trivial vs. rest of the pipeline)
__global__ void scan_excl(const int* __restrict__ cnt, int* __restrict__ off, int n) {
  __shared__ int sh[1024];
  __shared__ int carry;
  if (threadIdx.x == 0) carry = 0;
  __syncthreads();
  for (int base = 0; base < n; base += 1024) {
    int i = base + (int)threadIdx.x;
    int v = (i < n) ? cnt[i] : 0;
    sh[threadIdx.x] = v;
    __syncthreads();
    for (int s = 1; s < 1024; s <<= 1) {
      int t = (threadIdx.x >= (unsigned)s) ? sh[threadIdx.x - s] : 0;
      __syncthreads();
      sh[threadIdx.x] += t;
      __syncthreads();
    }
    if (i < n) off[i] = carry + sh[threadIdx.x] - v;  // exclusive
    __syncthreads();
    if (threadIdx.x == 1023) carry += sh[1023];
    __syncthreads();
  }
  if (threadIdx.x == 0) off[n] = carry;
}

__global__ void fill_edges(const int* __restrict__ src, const int* __restrict__ dst,
                           const int* __restrict__ off, int* __restrict__ fill,
                           int* __restrict__ eidx, int nE) {
  int e = blockIdx.x * blockDim.x + threadIdx.x;
  if (e < nE) {
    int d = dst[e];
    int p = off[d] + atomicAdd(&fill[d], 1);
    eidx[p] = src[e];
  }
}

// Pack [W_self; W_neigh] (2*dIn x N) into pair-interleaved, zero-padded Wp:
//   Wp[((r>>1)*Npad + c)*2 + (r&1)] = Wc[r][c]
// so a B fragment (rows 2k,2k+1 at column n) is ONE aligned b64 load.
__global__ void pack_weights(const float* __restrict__ Ws, const float* __restrict__ Wn,
                             const float* __restrict__ b, float* __restrict__ Wp,
                             float* __restrict__ bp, int dIn, int N, int Npad) {
  int idx = blockIdx.x * blockDim.x + threadIdx.x;
  int total = 2 * dIn * Npad;
  if (idx < total) {
    int r = idx / Npad;
    int c = idx - r * Npad;
    float v = 0.0f;
    if (c < N) v = (r < dIn) ? Ws[r * N + c] : Wn[(r - dIn) * N + c];
    Wp[((size_t)(r >> 1) * Npad + c) * 2 + (r & 1)] = v;
  }
  if (idx < Npad) bp[idx] = (idx < N) ? b[idx] : 0.0f;
}

// one block (d/4 lanes) per destination row; each lane owns a float4 feature chunk.
// Each edge contributes a fully coalesced d*4-byte row read; no atomics.
// Writes h_neigh already divided by max(deg,1). blockDim.x == d/4 (32 or 64).
__global__ void aggregate_mean(const float* __restrict__ h, const int* __restrict__ off,
                               const int* __restrict__ eidx, float* __restrict__ hneigh,
                               int d) {
  int dv  = blockIdx.x;
  int beg = off[dv];
  int end = off[dv + 1];
  float inv = 1.0f / fmaxf((float)(end - beg), 1.0f);
  int c4 = threadIdx.x * 4;
  float4 a0 = make_float4(0.f, 0.f, 0.f, 0.f);
  float4 a1 = make_float4(0.f, 0.f, 0.f, 0.f);
  int j = beg;
  for (; j + 1 < end; j += 2) {          // 2-way MLP on the dominant gather
    float4 u = *(const float4*)&h[(size_t)eidx[j]     * (size_t)d + (size_t)c4];
    float4 w = *(const float4*)&h[(size_t)eidx[j + 1] * (size_t)d + (size_t)c4];
    a0.x += u.x; a0.y += u.y; a0.z += u.z; a0.w += u.w;
    a1.x += w.x; a1.y += w.y; a1.z += w.z; a1.w += w.w;
  }
  if (j < end) {
    float4 u = *(const float4*)&h[(size_t)eidx[j] * (size_t)d + (size_t)c4];
    a0.x += u.x; a0.y += u.y; a0.z += u.z; a0.w += u.w;
  }
  float4 r = make_float4((a0.x + a1.x) * inv, (a0.y + a1.y) * inv,
                         (a0.z + a1.z) * inv, (a0.w + a1.w) * inv);
  *(float4*)&hneigh[(size_t)dv * (size_t)d + (size_t)c4] = r;
}

// ------------------------------- WMMA GEMM ---------------------------------
// out[M,NOUT] = [Hself | Hneigh] (M x 2*DIN) @ W (2*DIN x NPAD) + bp
// 256 threads = 8 wave32s; block owns a 16-row A tile in LDS and all NPAD
// columns. TWO=true: wave w computes tiles w and w+8 (two WMMA chains).
// A frag (16x4 f32): lane m=lane&15, k = k0 + 2*(lane>>4) + {0,1}
// B frag (4x16 f32): lane n=lane&15, same k pair (pair-packed -> b64 load)
// D (16x16 f32):     VGPR j -> row j + 8*(lane>>4), col = lane&15
template <int DIN, int NPAD, int NOUT, bool TWO, bool RELU>
__global__ __launch_bounds__(256)
void sage_gemm(const float* __restrict__ Hself, const float* __restrict__ Hneigh,
               const float* __restrict__ Wp, const float* __restrict__ bp,
               float* __restrict__ out) {
  constexpr int K   = 2 * DIN;
  constexpr int ldA = K + 4;             // +4 floats: bank skew, keeps 16B align
  __shared__ float Als[16 * ldA];

  const size_t mbase = (size_t)blockIdx.x * 16;

  // cooperative A-tile load, float4 granularity (DIN % 4 == 0: chunks never
  // straddle the self/neigh boundary)
  {
    constexpr int C4 = K / 4;            // float4 chunks per row
    for (int idx = threadIdx.x; idx < 16 * C4; idx += 256) {
      int r  = idx / C4;
      int k4 = (idx - r * C4) * 4;
      size_t row = mbase + (size_t)r;
      float4 v = (k4 < DIN)
          ? *(const float4*)&Hself [row * DIN + k4]
          : *(const float4*)&Hneigh[row * DIN + (k4 - DIN)];
      *(float4*)&Als[r * ldA + k4] = v;
    }
  }
  __syncthreads();

  const int wave = threadIdx.x >> 5;
  const int lane = threadIdx.x & 31;
  const int ml   = lane & 15;
  const int hi   = lane >> 4;
  const int arow = ml * ldA;

  const int n0 = wave * 16 + ml;         // tile `wave`
  const int n1 = n0 + 128;               // tile `wave + 8` (TWO only)
  const float* w0 = Wp + (size_t)n0 * 2; // pair-packed: row-pair rp at w[rp*NPAD*2]
  const float* w1 = Wp + (size_t)n1 * 2;

  v8f acc0 = {};
  v8f acc1 = {};
#pragma unroll 8
  for (int k0 = 0; k0 < K; k0 += 4) {
    const int rp = (k0 >> 1) + hi;       // row-pair index: rows 2*rp, 2*rp+1
    v2f a;
    a.x = Als[arow + 2 * rp];
    a.y = Als[arow + 2 * rp + 1];
    v2f b0 = *(const v2f*)&w0[(size_t)rp * (2 * NPAD)];
    acc0 = __builtin_amdgcn_wmma_f32_16x16x4_f32(false, a, false, b0,
                                                 (short)0, acc0, false, false);
    if constexpr (TWO) {
      v2f b1 = *(const v2f*)&w1[(size_t)rp * (2 * NPAD)];
      acc1 = __builtin_amdgcn_wmma_f32_16x16x4_f32(false, a, false, b1,
                                                   (short)0, acc1, false, false);
    }
  }

  const size_t rowbase = mbase + (size_t)(hi * 8);
  if (n0 < NOUT) {
    const float bv = bp[n0];
    float* op = out + rowbase * NOUT + n0;
    for (int j = 0; j < 8; ++j) {
      float v = acc0[j] + bv;
      if (RELU) v = fmaxf(v, 0.0f);
      op[j * NOUT] = v;
    }
  }
  if constexpr (TWO) {
    if (n1 < NOUT) {
      const float bv = bp[n1];
      float* op = out + rowbase * NOUT + n1;
      for (int j = 0; j < 8; ++j) {
        float v = acc1[j] + bv;
        if (RELU) v = fmaxf(v, 0.0f);
        op[j * NOUT] = v;
      }
    }
  }
}

// ------------------------------- launcher ----------------------------------

extern "C" void kernel_launch(void* const* d_in, const int* in_sizes, int n_in,
                              void* d_out, int out_size, void* d_ws, size_t ws_size,
                              hipStream_t stream) {
  (void)in_sizes; (void)n_in; (void)out_size; (void)ws_size;

  const float* x   = (const float*)d_in[0];
  const int* e0s = (const int*)d_in[1];  const int* e0d = (const int*)d_in[2];
  const int* e1s = (const int*)d_in[3];  const int* e1d = (const int*)d_in[4];
  const int* e2s = (const int*)d_in[5];  const int* e2d = (const int*)d_in[6];
  const float* Ws0 = (const float*)d_in[7];  const float* Wn0 = (const float*)d_in[8];
  const float* b0  = (const float*)d_in[9];
  const float* Ws1 = (const float*)d_in[10]; const float* Wn1 = (const float*)d_in[11];
  const float* b1  = (const float*)d_in[12];
  const float* Ws2 = (const float*)d_in[13]; const float* Wn2 = (const float*)d_in[14];
  const float* b2  = (const float*)d_in[15];
  float* out = (float*)d_out;

  // ---- workspace layout (floats first, then ints) ----
  float* f = (float*)d_ws;
  float* hneigh0 = f;  f += (size_t)N_DST0 * F_IN;    // 17.3M
  float* h1      = f;  f += (size_t)N_DST0 * F_HID;   // 34.6M
  float* hneigh1 = f;  f += (size_t)N_DST1 * F_HID;   //  3.1M
  float* h2      = f;  f += (size_t)N_DST1 * F_HID;   //  3.1M
  float* hneigh2 = f;  f += (size_t)N_DST2 * F_HID;   //  0.5M
  float* Wp0 = f;      f += (size_t)2 * F_IN  * F_HID;    // 256x256
  float* Wp1 = f;      f += (size_t)2 * F_HID * F_HID;    // 512x256
  float* Wp2 = f;      f += (size_t)2 * F_HID * F_OUT2P;  // 512x128
  float* bp0 = f;      f += F_HID;
  float* bp1 = f;      f += F_HID;
  float* bp2 = f;      f += F_OUT2P;
  int* ip = (int*)f;
  int* cnt0  = ip; ip += N_DST0;
  int* fill0 = ip; ip += N_DST0;
  int* cnt1  = ip; ip += N_DST1;
  int* fill1 = ip; ip += N_DST1;
  int* cnt2  = ip; ip += N_DST2;
  int* fill2 = ip; ip += N_DST2;
  const int nZero = 2 * (N_DST0 + N_DST1 + N_DST2);   // contiguous counter region
  int* off0 = ip; ip += N_DST0 + 1;
  int* off1 = ip; ip += N_DST1 + 1;
  int* off2 = ip; ip += N_DST2 + 1;
  int* eidx0 = ip; ip += N_E0;
  int* eidx1 = ip; ip += N_E1;
  int* eidx2 = ip; ip += N_E2;

  // counters must be zero every call (graph replays don't re-poison)
  zero_ints<<<(nZero + 255) / 256, 256, 0, stream>>>(cnt0, nZero);

  // pack weights (branch-free GEMM inner loops); tiny, cache-resident
  pack_weights<<<(2 * F_IN  * F_HID   + 255) / 256, 256, 0, stream>>>(Ws0, Wn0, b0, Wp0, bp0, F_IN,  F_HID, F_HID);
  pack_weights<<<(2 * F_HID * F_HID   + 255) / 256, 256, 0, stream>>>(Ws1, Wn1, b1, Wp1, bp1, F_HID, F_HID, F_HID);
  pack_weights<<<(2 * F_HID * F_OUT2P + 255) / 256, 256, 0, stream>>>(Ws2, Wn2, b2, Wp2, bp2, F_HID, F_OUT, F_OUT2P);

  // ---------------- layer 0: 128 -> 256, relu ----------------
  count_edges<<<(N_E0 + 255) / 256, 256, 0, stream>>>(e0d, cnt0, N_E0);
  scan_excl<<<1, 1024, 0, stream>>>(cnt0, off0, N_DST0);
  fill_edges<<<(N_E0 + 255) / 256, 256, 0, stream>>>(e0s, e0d, off0, fill0, eidx0, N_E0);
  aggregate_mean<<<N_DST0, F_IN / 4, 0, stream>>>(x, off0, eidx0, hneigh0, F_IN);
  sage_gemm<F_IN, F_HID, F_HID, true, true>
      <<<N_DST0 / 16, 256, 0, stream>>>(x, hneigh0, Wp0, bp0, h1);

  // ---------------- layer 1: 256 -> 256, relu ----------------
  count_edges<<<(N_E1 + 255) / 256, 256, 0, stream>>>(e1d, cnt1, N_E1);
  scan_excl<<<1, 1024, 0, stream>>>(cnt1, off1, N_DST1);
  fill_edges<<<(N_E1 + 255) / 256, 256, 0, stream>>>(e1s, e1d, off1, fill1, eidx1, N_E1);
  aggregate_mean<<<N_DST1, F_HID / 4, 0, stream>>>(h1, off1, eidx1, hneigh1, F_HID);
  sage_gemm<F_HID, F_HID, F_HID, true, true>
      <<<N_DST1 / 16, 256, 0, stream>>>(h1, hneigh1, Wp1, bp1, h2);

  // ---------------- layer 2: 256 -> 47, no relu ----------------
  count_edges<<<(N_E2 + 255) / 256, 256, 0, stream>>>(e2d, cnt2, N_E2);
  scan_excl<<<1, 1024, 0, stream>>>(cnt2, off2, N_DST2);
  fill_edges<<<(N_E2 + 255) / 256, 256, 0, stream>>>(e2s, e2d, off2, fill2, eidx2, N_E2);
  aggregate_mean<<<N_DST2, F_HID / 4, 0, stream>>>(h2, off2, eidx2, hneigh2, F_HID);
  sage_gemm<F_HID, F_OUT2P, F_OUT, false, false>
      <<<N_DST2 / 16, 256, 0, stream>>>(h2, hneigh2, Wp2, bp2, out);
}